// FTDNNPronscorer_64690797412607
// MI455X (gfx1250) — compile-verified
//
#include <hip/hip_runtime.h>

typedef __attribute__((ext_vector_type(2))) float v2f;
typedef __attribute__((ext_vector_type(8))) float v8f;

#define B_    32
#define T_    8192
#define DIN   256
#define OUT_  40
#define P_    800

#define NPAD        48          // OUT padded to 3 N-tiles of 16
#define LDS_STRIDE  260         // 256 + 4 pad to avoid LDS bank conflicts
#define WAVES       8
#define ROWS_PER_BLOCK (16 * WAVES)

// ---------------------------------------------------------------------------
// Kernel 1: masked projection via fp32 WMMA.
//   masked[bt, o] = (x[bt, :] . W[:, o] + bias[o]) * |mask[bt, o]|
// One wave computes a 16x48 output tile with V_WMMA_F32_16X16X4_F32,
// accumulating over K=256 in 64 steps of K=4.
// ---------------------------------------------------------------------------
__global__ __launch_bounds__(256)
void proj_mask_kernel(const float* __restrict__ x,
                      const float* __restrict__ W,
                      const float* __restrict__ bias,
                      const float* __restrict__ mask,
                      float* __restrict__ masked)
{
    __shared__ float ldsWt[NPAD * LDS_STRIDE];   // W transposed: [n][k], n padded to 48

    const int tid = threadIdx.x;

    // Stage W^T (zero-padded columns 40..47) into LDS, cooperatively.
    for (int i = tid; i < NPAD * DIN; i += 256) {
        const int n = i / DIN;
        const int k = i - n * DIN;
        const float v = (n < OUT_) ? W[k * OUT_ + n] : 0.0f;
        ldsWt[n * LDS_STRIDE + k] = v;
    }
    __syncthreads();

    const int wave  = tid >> 5;
    const int lane  = tid & 31;
    const int lm    = lane & 15;           // M row (A) / N col (B,C) within tile
    const int khalf = (lane >> 4) << 1;    // lanes 0-15 hold K={0,1}, 16-31 hold K={2,3}

    const long row_base = (long)blockIdx.x * ROWS_PER_BLOCK + wave * 16;

    const float* arow = x + (row_base + lm) * DIN + khalf;
    const float* b0p  = &ldsWt[( 0 + lm) * LDS_STRIDE + khalf];
    const float* b1p  = &ldsWt[(16 + lm) * LDS_STRIDE + khalf];
    const float* b2p  = &ldsWt[(32 + lm) * LDS_STRIDE + khalf];

    v8f c0 = {}, c1 = {}, c2 = {};

#pragma unroll 4
    for (int k = 0; k < DIN; k += 4) {
        v2f a  = *(const v2f*)(arow + k);
        v2f b0 = *(const v2f*)(b0p + k);
        v2f b1 = *(const v2f*)(b1p + k);
        v2f b2 = *(const v2f*)(b2p + k);
        // 8 args: (neg_a, A, neg_b, B, c_mod, C, reuse_a, reuse_b)
        c0 = __builtin_amdgcn_wmma_f32_16x16x4_f32(false, a, false, b0, (short)0, c0, false, false);
        c1 = __builtin_amdgcn_wmma_f32_16x16x4_f32(false, a, false, b1, (short)0, c1, false, false);
        c2 = __builtin_amdgcn_wmma_f32_16x16x4_f32(false, a, false, b2, (short)0, c2, false, false);
    }

    // C/D layout: VGPR r holds (M = r + 8*(lane/16), N = lane%16).
    const int mbase = (lane >> 4) * 8;
    const int n0 = lm;
    const int n1 = 16 + lm;
    const int n2 = 32 + lm;     // only stored when < OUT_
#pragma unroll
    for (int r = 0; r < 8; ++r) {
        const long row = row_base + mbase + r;
        const long ob  = row * OUT_;
        float v0 = c0[r] + bias[n0];
        float v1 = c1[r] + bias[n1];
        masked[ob + n0] = v0 * fabsf(mask[ob + n0]);
        masked[ob + n1] = v1 * fabsf(mask[ob + n1]);
        if (n2 < OUT_) {
            float v2 = c2[r] + bias[n2];
            masked[ob + n2] = v2 * fabsf(mask[ob + n2]);
        }
    }
}

// ---------------------------------------------------------------------------
// Kernel 2: per-phone segment means.
// cumsum-gather-diff over sorted end indices == direct segment sums over
// t in (idx1[p-1], idx1[p]].  counts from the *signed* mask, clamped to 1.
// One block per (b, p); threadIdx.x = output channel.
// ---------------------------------------------------------------------------
__global__ __launch_bounds__(64)
void segmean_kernel(const float* __restrict__ masked,
                    const float* __restrict__ mask,
                    const int* __restrict__ idx0,
                    const int* __restrict__ idx1,
                    float* __restrict__ out)
{
    const int bp = blockIdx.x;
    const int b  = bp / P_;
    const int p  = bp - b * P_;
    const int o  = threadIdx.x;
    if (o >= OUT_) return;

    const int ib    = idx0[b * P_ + p];                    // utterance index (== b)
    const int end   = idx1[b * P_ + p];                    // inclusive end frame
    const int start = (p > 0) ? idx1[b * P_ + p - 1] : -1; // previous segment end

    float s = 0.0f, cnt = 0.0f;
    for (int t = start + 1; t <= end; ++t) {
        const long base = ((long)ib * T_ + t) * OUT_ + o;
        s   += masked[base];
        cnt += mask[base];
    }
    if (cnt == 0.0f) cnt = 1.0f;
    out[((long)b * P_ + p) * OUT_ + o] = s / cnt;
}

// ---------------------------------------------------------------------------
extern "C" void kernel_launch(void* const* d_in, const int* in_sizes, int n_in,
                              void* d_out, int out_size, void* d_ws, size_t ws_size,
                              hipStream_t stream)
{
    const float* x    = (const float*)d_in[0];   // [B,T,DIN]
    const float* W    = (const float*)d_in[1];   // [DIN,OUT]
    const float* bias = (const float*)d_in[2];   // [OUT]
    const float* mask = (const float*)d_in[3];   // [B,T,OUT] signed phone mask
    const int*   bidx = (const int*)d_in[4];     // [2,B,P]
    // d_in[5..7]: loss_per_phone=1, evaluation=0, normalize=1 (fixed path)

    float* ws_masked = (float*)d_ws;             // B*T*OUT floats = 41.9 MB

    const int*   idx0 = bidx;                    // [B,P]
    const int*   idx1 = bidx + B_ * P_;          // [B,P] sorted end frames

    const int grid1 = (B_ * T_) / ROWS_PER_BLOCK; // 2048
    proj_mask_kernel<<<grid1, 256, 0, stream>>>(x, W, bias, mask, ws_masked);

    segmean_kernel<<<B_ * P_, 64, 0, stream>>>(ws_masked, mask, idx0, idx1,
                                               (float*)d_out);
}